// SampleGNN_16355235463625
// MI455X (gfx1250) — compile-verified
//
#include <hip/hip_runtime.h>

typedef __attribute__((ext_vector_type(2))) float v2f;
typedef __attribute__((ext_vector_type(8))) float v8f;

#define HDIM 64

// ---------------- degree / norm ----------------

__global__ void k_init_deg(float* deg, int n) {
    int i = blockIdx.x * blockDim.x + threadIdx.x;
    if (i < n) deg[i] = 1.0f;  // self loop
}

__global__ void k_deg_edges(const int* __restrict__ dst, float* deg, int nE) {
    int e = blockIdx.x * blockDim.x + threadIdx.x;
    if (e < nE) unsafeAtomicAdd(deg + dst[e], 1.0f);
}

__global__ void k_finalize_dinv(float* dinv, int n) {
    int i = blockIdx.x * blockDim.x + threadIdx.x;
    if (i < n) dinv[i] = rsqrtf(dinv[i]);
}

__global__ void k_edge_norm(const int* __restrict__ src, const int* __restrict__ dst,
                            const float* __restrict__ dinv, float* __restrict__ norm, int nE) {
    int e = blockIdx.x * blockDim.x + threadIdx.x;
    if (e < nE) norm[e] = dinv[src[e]] * dinv[dst[e]];
}

// ---------------- x @ W1 (K=5, too skinny for WMMA) ----------------

__global__ void k_xw1(const float* __restrict__ x, const float* __restrict__ W1,
                      float* __restrict__ t1, int n) {
    int tid = blockIdx.x * blockDim.x + threadIdx.x;
    int v = tid >> 6, f = tid & 63;
    if (v >= n) return;
    const float* xr = x + (size_t)v * 5;
    float s = 0.f;
#pragma unroll
    for (int k = 0; k < 5; ++k) s = fmaf(xr[k], W1[k * HDIM + f], s);
    t1[tid] = s;
}

// ---------------- agg init: self-loop contribution + bias ----------------

__global__ void k_agg_init(const float* __restrict__ t, const float* __restrict__ dinv,
                           const float* __restrict__ bias, float* __restrict__ agg, int n) {
    int tid = blockIdx.x * blockDim.x + threadIdx.x;
    int v = tid >> 6, f = tid & 63;
    if (v >= n) return;
    float di = dinv[v];
    agg[tid] = t[tid] * di * di + bias[f];
}

// ---------------- edge scatter-add: agg[dst] += t[src] * norm ----------------

__global__ void k_scatter(const float* __restrict__ t, const float* __restrict__ norm,
                          const int* __restrict__ src, const int* __restrict__ dst,
                          float* __restrict__ agg, int nE) {
    int tid = blockIdx.x * blockDim.x + threadIdx.x;
    int e = tid >> 4, c = tid & 15;  // 16 float4 chunks cover 64 feats
    if (e >= nE) return;
    int s = src[e], d = dst[e];
    float w = norm[e];
    const float4 v = *(const float4*)(t + (size_t)s * HDIM + c * 4);
    float* p = agg + (size_t)d * HDIM + c * 4;
    unsafeAtomicAdd(p + 0, v.x * w);
    unsafeAtomicAdd(p + 1, v.y * w);
    unsafeAtomicAdd(p + 2, v.z * w);
    unsafeAtomicAdd(p + 3, v.w * w);
}

// ---------------- t2 = relu(agg1) @ W2 via V_WMMA_F32_16X16X4_F32 ----------------
// Block = 128 threads = 4 waves; each wave computes one 16-row x 64-col strip.
// A frag (16x4 f32): lanes 0-15 hold rows M=0..15 K={kb,kb+1}; lanes 16-31 hold K={kb+2,kb+3}.
// B frag (4x16 f32): N striped across lanes 0-15 (mirrored for lanes 16-31), K split like A.

__global__ void k_gemm_relu_wmma(const float* __restrict__ A, const float* __restrict__ W,
                                 float* __restrict__ Out, int n) {
    __shared__ float sW[HDIM * HDIM];  // 16 KB of the 320 KB/WGP LDS
    for (int i = threadIdx.x * 4; i < HDIM * HDIM; i += blockDim.x * 4)
        *(float4*)(sW + i) = *(const float4*)(W + i);
    __syncthreads();

    const int wave = threadIdx.x >> 5;
    const int lane = threadIdx.x & 31;
    const int mtile = blockIdx.x * 4 + wave;
    const int row0 = mtile * 16;
    if (row0 >= n) return;  // wave-uniform exit: EXEC stays all-ones inside live waves

    const int mrow  = lane & 15;
    const int khalf = lane >> 4;  // 0: K={kb,kb+1}, 1: K={kb+2,kb+3}
    const int row   = min(row0 + mrow, n - 1);
    const float* __restrict__ arow = A + (size_t)row * HDIM;

    v8f acc0 = {}, acc1 = {}, acc2 = {}, acc3 = {};

#pragma unroll
    for (int kt = 0; kt < 16; ++kt) {
        const int kb = kt * 4 + khalf * 2;
        v2f a;
        a.x = fmaxf(arow[kb + 0], 0.f);  // fused ReLU on the fly
        a.y = fmaxf(arow[kb + 1], 0.f);

        v2f b0, b1, b2, b3;
        b0.x = sW[(kb + 0) * HDIM + ( 0 + mrow)]; b0.y = sW[(kb + 1) * HDIM + ( 0 + mrow)];
        b1.x = sW[(kb + 0) * HDIM + (16 + mrow)]; b1.y = sW[(kb + 1) * HDIM + (16 + mrow)];
        b2.x = sW[(kb + 0) * HDIM + (32 + mrow)]; b2.y = sW[(kb + 1) * HDIM + (32 + mrow)];
        b3.x = sW[(kb + 0) * HDIM + (48 + mrow)]; b3.y = sW[(kb + 1) * HDIM + (48 + mrow)];

        acc0 = __builtin_amdgcn_wmma_f32_16x16x4_f32(false, a, false, b0, (short)0, acc0, false, false);
        acc1 = __builtin_amdgcn_wmma_f32_16x16x4_f32(false, a, false, b1, (short)0, acc1, false, false);
        acc2 = __builtin_amdgcn_wmma_f32_16x16x4_f32(false, a, false, b2, (short)0, acc2, false, false);
        acc3 = __builtin_amdgcn_wmma_f32_16x16x4_f32(false, a, false, b3, (short)0, acc3, false, false);
    }

    // D layout: VGPR r holds M=r (lanes 0-15) / M=r+8 (lanes 16-31), N=lane%16.
#pragma unroll
    for (int r = 0; r < 8; ++r) {
        const int m = r + khalf * 8;
        if (row0 + m < n) {
            float* orow = Out + (size_t)(row0 + m) * HDIM + mrow;
            orow[ 0] = acc0[r];
            orow[16] = acc1[r];
            orow[32] = acc2[r];
            orow[48] = acc3[r];
        }
    }
}

// ---------------- global mean pool (sorted batch -> L2-local atomics) ----------------

__global__ void k_pool_zero(float* pooled, float* counts, int g) {
    int i = blockIdx.x * blockDim.x + threadIdx.x;
    if (i < g * HDIM) pooled[i] = 0.f;
    if (i < g) counts[i] = 0.f;
}

__global__ void k_pool_scatter(const float* __restrict__ agg, const int* __restrict__ batch,
                               float* __restrict__ pooled, float* __restrict__ counts, int n) {
    int tid = blockIdx.x * blockDim.x + threadIdx.x;
    int v = tid >> 4, c = tid & 15;
    if (v >= n) return;
    int g = batch[v];
    const float4 h = *(const float4*)(agg + (size_t)v * HDIM + c * 4);  // relu below
    float* p = pooled + (size_t)g * HDIM + c * 4;
    unsafeAtomicAdd(p + 0, fmaxf(h.x, 0.f));
    unsafeAtomicAdd(p + 1, fmaxf(h.y, 0.f));
    unsafeAtomicAdd(p + 2, fmaxf(h.z, 0.f));
    unsafeAtomicAdd(p + 3, fmaxf(h.w, 0.f));
    if (c == 0) unsafeAtomicAdd(counts + g, 1.0f);
}

// ---------------- MLP head: z = relu(mean @ Wh1 + bh1); out = z @ Wh2 + bh2 ----------------

__global__ void k_head(const float* __restrict__ pooled, const float* __restrict__ counts,
                       const float* __restrict__ Wh1, const float* __restrict__ bh1,
                       const float* __restrict__ Wh2, const float* __restrict__ bh2,
                       float* __restrict__ out, int g) {
    int gi = blockIdx.x * blockDim.x + threadIdx.x;
    if (gi >= g) return;
    float inv = 1.0f / fmaxf(counts[gi], 1.0f);
    float m[HDIM];
#pragma unroll
    for (int k = 0; k < HDIM; ++k) m[k] = pooled[(size_t)gi * HDIM + k] * inv;
    float o0 = bh2[0], o1 = bh2[1];
    for (int j = 0; j < 32; ++j) {
        float s = bh1[j];
#pragma unroll
        for (int k = 0; k < HDIM; ++k) s = fmaf(m[k], Wh1[k * 32 + j], s);
        s = fmaxf(s, 0.f);
        o0 = fmaf(s, Wh2[j * 2 + 0], o0);
        o1 = fmaf(s, Wh2[j * 2 + 1], o1);
    }
    out[gi * 2 + 0] = o0;
    out[gi * 2 + 1] = o1;
}

// ---------------- launch ----------------

static inline int cdiv(long long a, int b) { return (int)((a + b - 1) / b); }

extern "C" void kernel_launch(void* const* d_in, const int* in_sizes, int n_in,
                              void* d_out, int out_size, void* d_ws, size_t ws_size,
                              hipStream_t stream) {
    const float* x    = (const float*)d_in[0];
    const int*   ei   = (const int*)d_in[1];
    const int*   batch= (const int*)d_in[2];
    const float* W1   = (const float*)d_in[3];
    const float* b1   = (const float*)d_in[4];
    const float* W2   = (const float*)d_in[5];
    const float* b2   = (const float*)d_in[6];
    const float* Wh1  = (const float*)d_in[7];
    const float* bh1  = (const float*)d_in[8];
    const float* Wh2  = (const float*)d_in[9];
    const float* bh2  = (const float*)d_in[10];
    float* out = (float*)d_out;

    const int N = in_sizes[2];
    const int E = in_sizes[1] / 2;
    const int G = out_size / 2;
    const int* src = ei;
    const int* dst = ei + E;

    // workspace carve-up (floats), regions 64-element aligned for float4 access
    float* ws = (float*)d_ws;
    size_t off = 0;
    float* dinv   = ws + off; off += ((size_t)N + 63) & ~(size_t)63;
    float* norm   = ws + off; off += ((size_t)E + 63) & ~(size_t)63;
    float* bufT   = ws + off; off += (size_t)N * HDIM;  // t1, later t2
    float* bufA   = ws + off; off += (size_t)N * HDIM;  // agg1, later agg2
    float* pooled = ws + off; off += (size_t)G * HDIM;
    float* counts = ws + off; off += (size_t)G;
    (void)ws_size; (void)n_in;

    const int B = 256;

    // degrees -> dinv -> per-edge norm
    k_init_deg<<<cdiv(N, B), B, 0, stream>>>(dinv, N);
    k_deg_edges<<<cdiv(E, B), B, 0, stream>>>(dst, dinv, E);
    k_finalize_dinv<<<cdiv(N, B), B, 0, stream>>>(dinv, N);
    k_edge_norm<<<cdiv(E, B), B, 0, stream>>>(src, dst, dinv, norm, E);

    // conv1: t1 = x@W1 ; agg1 = selfloop+bias ; scatter edges
    k_xw1<<<cdiv((long long)N * HDIM, B), B, 0, stream>>>(x, W1, bufT, N);
    k_agg_init<<<cdiv((long long)N * HDIM, B), B, 0, stream>>>(bufT, dinv, b1, bufA, N);
    k_scatter<<<cdiv((long long)E * 16, B), B, 0, stream>>>(bufT, norm, src, dst, bufA, E);

    // conv2: t2 = relu(agg1)@W2 (WMMA) ; agg2 ; scatter
    {
        int tiles = cdiv(N, 16);
        k_gemm_relu_wmma<<<cdiv(tiles, 4), 128, 0, stream>>>(bufA, W2, bufT, N);
    }
    k_agg_init<<<cdiv((long long)N * HDIM, B), B, 0, stream>>>(bufT, dinv, b2, bufA, N);
    k_scatter<<<cdiv((long long)E * 16, B), B, 0, stream>>>(bufT, norm, src, dst, bufA, E);

    // global mean pool (relu fused into scatter)
    k_pool_zero<<<cdiv((long long)G * HDIM, B), B, 0, stream>>>(pooled, counts, G);
    k_pool_scatter<<<cdiv((long long)N * 16, B), B, 0, stream>>>(bufA, batch, pooled, counts, N);

    // MLP head
    k_head<<<cdiv(G, 128), 128, 0, stream>>>(pooled, counts, Wh1, bh1, Wh2, bh2, out, G);
}